// GraphAutoencoder_50749333569689
// MI455X (gfx1250) — compile-verified
//
#include <hip/hip_runtime.h>

typedef __attribute__((ext_vector_type(2))) float v2f;
typedef __attribute__((ext_vector_type(8))) float v8f;

constexpr int HD = 128;   // hidden/latent dim

// ---------------- Tensor Data Mover staging (CDNA5 TDM) ----------------
#if defined(__has_builtin)
#if __has_builtin(__builtin_amdgcn_tensor_load_to_lds)
#define USE_TDM 1
#endif
#endif

#ifdef USE_TDM
typedef unsigned int u32;
typedef __attribute__((ext_vector_type(4))) unsigned int u32x4;
typedef __attribute__((ext_vector_type(4))) int i32x4;
typedef __attribute__((ext_vector_type(8))) int i32x8;

// DMA a [rows x 128] fp32 slab (row-major, stride 128) from global to LDS.
// Descriptor per CDNA5 ISA 8.3/8.4: group0 = {flags, lds_addr, global_addr, type},
// group1 = {data_size, tensor dims, tile dims, dim0 stride}. 2D tile; groups 2/3 zero.
__device__ __forceinline__ void tdm_load_f32_rows(const float* gsrc, float* ldsDst, int rows) {
  unsigned long long ga = (unsigned long long)gsrc;
  u32 lds_off = (u32)(unsigned long long)ldsDst;   // generic LDS addr low 32 bits = LDS offset
  u32x4 g0;
  g0[0] = 1u;                                            // count=1, user descriptor
  g0[1] = lds_off;                                       // lds_addr (bytes)
  g0[2] = (u32)ga;                                       // global_addr[31:0]  (bits 95:64)
  g0[3] = (u32)((ga >> 32) & 0x01FFFFFFull) | (2u << 30);// global_addr[56:32] | type=2
  i32x8 g1;
  g1[0] = 2 << 16;          // workgroup_mask=0, data_size=2 (4 bytes)
  g1[1] = HD << 16;         // tensor_dim0 = 128        (bits 79:48, low half)
  g1[2] = rows << 16;       // tensor_dim1 = rows       (bits 111:80, low half)
  g1[3] = HD << 16;         // tile_dim0   = 128        (bits 127:112)
  g1[4] = rows;             // tile_dim1   = rows, tile_dim2 = 0
  g1[5] = HD;               // tensor_dim0_stride = 128 (bits 207:160, low 32)
  g1[6] = 0;                // stride hi / dim1 stride (unused, 2D)
  g1[7] = 0;
  i32x4 z4 = {0, 0, 0, 0};
#if __clang_major__ >= 23
  i32x8 z8 = {0, 0, 0, 0, 0, 0, 0, 0};
  __builtin_amdgcn_tensor_load_to_lds(g0, g1, z4, z4, z8, 0);
#else
  __builtin_amdgcn_tensor_load_to_lds(g0, g1, z4, z4, 0);
#endif
#if __has_builtin(__builtin_amdgcn_s_wait_tensorcnt)
  __builtin_amdgcn_s_wait_tensorcnt(0);
#else
  asm volatile("s_wait_tensorcnt 0x0" ::: "memory");
#endif
}
#endif  // USE_TDM

// ---------------- degree / normalization ----------------
__global__ __launch_bounds__(256) void k_fill_deg(float* __restrict__ deg, int n) {
  int i = blockIdx.x * 256 + threadIdx.x;
  if (i < n) deg[i] = 1.0f;               // self-loop contributes 1
}

__global__ __launch_bounds__(256) void k_deg_acc(const int* __restrict__ dst,
                                                 float* __restrict__ deg, int ne) {
  int i = blockIdx.x * 256 + threadIdx.x;
  if (i < ne)
    __hip_atomic_fetch_add(&deg[dst[i]], 1.0f, __ATOMIC_RELAXED, __HIP_MEMORY_SCOPE_AGENT);
}

__global__ __launch_bounds__(256) void k_rsqrt(float* __restrict__ d, int n) {
  int i = blockIdx.x * 256 + threadIdx.x;
  if (i < n) d[i] = rsqrtf(d[i]);         // deg >= 1 always (self loops)
}

// ---------------- conv1 linear (K=6, VALU) + self-loop init ----------------
__global__ __launch_bounds__(256) void k_xw1(const float* __restrict__ x,
                                             const float* __restrict__ W1,
                                             const float* __restrict__ dinv,
                                             float* __restrict__ xw,
                                             float* __restrict__ selfacc, int n) {
  int idx = blockIdx.x * 256 + threadIdx.x;
  if (idx >= n * HD) return;
  int node = idx >> 7, f = idx & (HD - 1);
  float acc = 0.0f;
#pragma unroll
  for (int i = 0; i < 6; ++i)
    acc = fmaf(x[node * 6 + i], W1[i * HD + f], acc);
  xw[idx] = acc;
  float di = dinv[node];
  selfacc[idx] = acc * di * di;           // self-loop term: norm = dinv[n]^2
}

// ---------------- self-loop init for conv2 ----------------
__global__ __launch_bounds__(256) void k_selfinit(const float* __restrict__ in,
                                                  const float* __restrict__ dinv,
                                                  float* __restrict__ out, int total) {
  int idx = blockIdx.x * 256 + threadIdx.x;
  if (idx >= total) return;
  float di = dinv[idx >> 7];
  out[idx] = in[idx] * di * di;
}

// ---------------- edge scatter: one wave per edge, float4 per lane ----------------
__global__ __launch_bounds__(256) void k_scatter(const float* __restrict__ xw,
                                                 const float* __restrict__ dinv,
                                                 const int* __restrict__ src,
                                                 const int* __restrict__ dst,
                                                 float* __restrict__ out, int ne) {
  int e = (blockIdx.x * 256 + threadIdx.x) >> 5;
  if (e >= ne) return;
  int lane = threadIdx.x & 31;
  int s = src[e], d = dst[e];
  float c = dinv[s] * dinv[d];
  float4 v = ((const float4*)(xw + (size_t)s * HD))[lane];   // 32 lanes * 4 = 128
  float* o = out + (size_t)d * HD + lane * 4;
  __hip_atomic_fetch_add(o + 0, v.x * c, __ATOMIC_RELAXED, __HIP_MEMORY_SCOPE_AGENT);
  __hip_atomic_fetch_add(o + 1, v.y * c, __ATOMIC_RELAXED, __HIP_MEMORY_SCOPE_AGENT);
  __hip_atomic_fetch_add(o + 2, v.z * c, __ATOMIC_RELAXED, __HIP_MEMORY_SCOPE_AGENT);
  __hip_atomic_fetch_add(o + 3, v.w * c, __ATOMIC_RELAXED, __HIP_MEMORY_SCOPE_AGENT);
}

// ---------------- elementwise bias (+ optional relu) ----------------
__global__ __launch_bounds__(256) void k_bias_act(float* __restrict__ buf,
                                                  const float* __restrict__ b,
                                                  int total, int relu) {
  int idx = blockIdx.x * 256 + threadIdx.x;
  if (idx >= total) return;
  float v = buf[idx] + b[idx & (HD - 1)];
  if (relu) v = fmaxf(v, 0.0f);
  buf[idx] = v;
}

// ---------------- WMMA GEMM: C[nrows x 128] = A[nrows x 128] @ B[128 x 128] ----------------
// block = 256 threads = 8 waves; block computes a 16-row x 128-col slab.
// B staged in 64KB LDS: each wave TDM-DMAs its own 16x128 slice (TDM ignores EXEC,
// so every wave issues unconditionally and drains its own TENSORcnt).
// Fragments per CDNA5 ISA 7.12.2:
//   A 16x4 f32: lane L -> M=L&15, v2f = A[M][k0+2*(L>>4) .. +1]
//   B 4x16 f32: lane L -> N=L&15, v2f = B[k0+2*(L>>4)][N], B[k0+2*(L>>4)+1][N]
//   C 16x16 f32: v8f, VGPR r -> M = r + 8*(L>>4), N = L&15
__global__ __launch_bounds__(256) void k_gemm128(const float* __restrict__ A,
                                                 const float* __restrict__ B,
                                                 const float* __restrict__ bias,
                                                 float* __restrict__ C,
                                                 int nrows, int addBias, int doRelu) {
  __shared__ float Bs[HD * HD];           // 64 KB
  int tid = threadIdx.x;
  int wave = tid >> 5, lane = tid & 31;

#ifdef USE_TDM
  // wave w DMAs B rows [16w, 16w+16) -> Bs (one TDM op per wave, in-order per wave)
  tdm_load_f32_rows(B + (size_t)wave * 16 * HD, Bs + wave * 16 * HD, 16);
  __syncthreads();
#else
  {
    const float4* bsrc = (const float4*)B;
    float4* bdst = (float4*)Bs;
#pragma unroll
    for (int i = 0; i < (HD * HD / 4) / 256; ++i)      // 16 iters
      bdst[tid + i * 256] = bsrc[tid + i * 256];
  }
  __syncthreads();
#endif

  int row0 = blockIdx.x * 16;
  int m = lane & 15, hi = lane >> 4;
  int col = wave * 16 + m;
  int arow_i = row0 + m;
  if (arow_i > nrows - 1) arow_i = nrows - 1;          // clamp (no divergence)
  const float* arow = A + (size_t)arow_i * HD;

  v8f acc = {};
#pragma unroll 4
  for (int k0 = 0; k0 < HD; k0 += 4) {
    int ka = k0 + 2 * hi;
    v2f a = *(const v2f*)(arow + ka);                  // 8B aligned (ka even)
    v2f b;
    b.x = Bs[ka * HD + col];
    b.y = Bs[(ka + 1) * HD + col];
    acc = __builtin_amdgcn_wmma_f32_16x16x4_f32(
        /*neg_a=*/false, a, /*neg_b=*/false, b,
        /*c_mod=*/(short)0, acc, /*reuse_a=*/false, /*reuse_b=*/false);
  }

  float bv = addBias ? bias[col] : 0.0f;
  if (row0 + 16 <= nrows) {                            // block-uniform fast path
    float* cb = C + (size_t)(row0 + 8 * hi) * HD + col;
#pragma unroll
    for (int r = 0; r < 8; ++r) {
      float v = acc[r] + bv;
      if (doRelu) v = fmaxf(v, 0.0f);
      cb[(size_t)r * HD] = v;
    }
  } else {
#pragma unroll
    for (int r = 0; r < 8; ++r) {
      int rr = row0 + r + 8 * hi;
      float v = acc[r] + bv;
      if (doRelu) v = fmaxf(v, 0.0f);
      if (rr < nrows) C[(size_t)rr * HD + col] = v;
    }
  }
}

// ---------------- final decoder layer: [N x 128] @ [128 x 6] + b ----------------
__global__ __launch_bounds__(256) void k_fc2(const float* __restrict__ A,
                                             const float* __restrict__ W,
                                             const float* __restrict__ b,
                                             float* __restrict__ out, int n) {
  int t = blockIdx.x * 256 + threadIdx.x;
  int node = t >> 3, j = t & 7;
  if (node >= n || j >= 6) return;
  float acc = b[j];
  const float* ar = A + (size_t)node * HD;
#pragma unroll 8
  for (int k = 0; k < HD; ++k)
    acc = fmaf(ar[k], W[k * 6 + j], acc);
  out[node * 6 + j] = acc;
}

// ---------------- host side ----------------
extern "C" void kernel_launch(void* const* d_in, const int* in_sizes, int n_in,
                              void* d_out, int out_size, void* d_ws, size_t ws_size,
                              hipStream_t stream) {
  const float* x    = (const float*)d_in[0];
  const int*   eidx = (const int*)  d_in[1];
  const float* W1   = (const float*)d_in[2];
  const float* b1   = (const float*)d_in[3];
  const float* W2   = (const float*)d_in[4];
  const float* b2   = (const float*)d_in[5];
  const float* fc1w = (const float*)d_in[6];
  const float* fc1b = (const float*)d_in[7];
  const float* fc2w = (const float*)d_in[8];
  const float* fc2b = (const float*)d_in[9];
  float* out = (float*)d_out;

  const int N = in_sizes[0] / 6;
  const int E = in_sizes[1] / 2;
  const int* src = eidx;        // edge_index[0]
  const int* dst = eidx + E;    // edge_index[1]

  float* ws   = (float*)d_ws;
  float* dinv = ws;                          // N floats
  float* buf0 = ws + N;                      // N*128: xw1 -> hw2 -> d1
  float* buf1 = buf0 + (size_t)N * HD;       // N*128: h   -> z

  auto cdiv = [](long a, long b) { return (int)((a + b - 1) / b); };
  const int T = 256;
  const long NH = (long)N * HD;

  // normalization coefficients
  k_fill_deg<<<cdiv(N, T), T, 0, stream>>>(dinv, N);
  k_deg_acc <<<cdiv(E, T), T, 0, stream>>>(dst, dinv, E);
  k_rsqrt   <<<cdiv(N, T), T, 0, stream>>>(dinv, N);

  // conv1: xw1 (buf0), self-term init (buf1), scatter, +b1 relu -> h (buf1)
  k_xw1     <<<cdiv(NH, T), T, 0, stream>>>(x, W1, dinv, buf0, buf1, N);
  k_scatter <<<cdiv((long)E * 32, T), T, 0, stream>>>(buf0, dinv, src, dst, buf1, E);
  k_bias_act<<<cdiv(NH, T), T, 0, stream>>>(buf1, b1, (int)NH, 1);

  // conv2: hw2 = h@W2 (buf0), self init (buf1), scatter, +b2 -> z (buf1)
  k_gemm128 <<<cdiv(N, 16), T, 0, stream>>>(buf1, W2, nullptr, buf0, N, 0, 0);
  k_selfinit<<<cdiv(NH, T), T, 0, stream>>>(buf0, dinv, buf1, (int)NH);
  k_scatter <<<cdiv((long)E * 32, T), T, 0, stream>>>(buf0, dinv, src, dst, buf1, E);
  k_bias_act<<<cdiv(NH, T), T, 0, stream>>>(buf1, b2, (int)NH, 0);

  // decoder: d1 = relu(z@fc1 + fc1_b) (buf0), out = d1@fc2 + fc2_b
  k_gemm128 <<<cdiv(N, 16), T, 0, stream>>>(buf1, fc1w, fc1b, buf0, N, 1, 1);
  k_fc2     <<<cdiv((long)N * 8, T), T, 0, stream>>>(buf0, fc2w, fc2b, out, N);
}